// Attention_7911329759940
// MI455X (gfx1250) — compile-verified
//
#include <hip/hip_runtime.h>
#include <math.h>

typedef float v2f __attribute__((ext_vector_type(2)));
typedef float v8f __attribute__((ext_vector_type(8)));

#define B_      256
#define T_      2048
#define RNN_    1024
#define EMB_    512
#define ATT_    128
#define NF_     32
#define KS_     31
#define KJ_     62      // 2*31 im2row K
#define KJP_    64      // padded to WMMA K multiple
#define TCHUNK_ 128
#define WPSTRIDE_ 136   // LDS row stride (in v2f pairs) for Weff pairs
#define NSPLIT_ 4

// workspace layout (in floats)
#define WS_PQ    0
#define WS_WEFF  (WS_PQ + B_*ATT_)            // 32768
#define WS_E     (WS_WEFF + KJP_*ATT_)        // 40960
#define WS_PART  (WS_E + B_*T_)               // 565248
// total floats: 565248 + NSPLIT_*B_*EMB_ = 1,089,536  (~4.4 MB)

// ---------------------------------------------------------------------------
// Kernel 0: fold conv weights into Weff^T[j][a] = sum_f Wloc[a,f]*conv_w[f,j]
// ---------------------------------------------------------------------------
__global__ void weff_kernel(const float* __restrict__ convw,   // [32,2,31]
                            const float* __restrict__ wloc,    // [128,32]
                            float* __restrict__ weff)          // [64,128]
{
    for (int idx = threadIdx.x; idx < KJP_*ATT_; idx += blockDim.x) {
        int j = idx >> 7, a = idx & (ATT_-1);
        float s = 0.f;
        if (j < KJ_) {
            #pragma unroll
            for (int f = 0; f < NF_; ++f)
                s += wloc[a*NF_ + f] * convw[f*KJ_ + j];
        }
        weff[idx] = s;
    }
}

// ---------------------------------------------------------------------------
// Kernel 1: pq[b,a] = sum_k H[b,k]*Wq[a,k]  via V_WMMA_F32_16X16X4_F32
// grid: B/16 blocks, 8 waves; wave w owns 16x16 tile (rows b0..b0+15, cols w*16..)
// ---------------------------------------------------------------------------
__global__ __launch_bounds__(256) void pq_kernel(const float* __restrict__ H,
                                                 const float* __restrict__ Wq,
                                                 float* __restrict__ pq)
{
    const int tid  = threadIdx.x;
    const int wave = tid >> 5, lane = tid & 31;
    const int h = lane >> 4, r = lane & 15;
    const int b0 = blockIdx.x * 16;
    const int n0 = wave * 16;

    v8f acc = {0,0,0,0,0,0,0,0};
    const v2f* Arow = (const v2f*)(H  + (size_t)(b0 + r) * RNN_ + 2*h); // K = k+2h, k+2h+1
    const v2f* Brow = (const v2f*)(Wq + (size_t)(n0 + r) * RNN_ + 2*h); // B (K x N): Wq[n,k]
    #pragma unroll 4
    for (int k = 0; k < RNN_/4; ++k) {
        v2f a = Arow[2*k];
        v2f b = Brow[2*k];
        acc = __builtin_amdgcn_wmma_f32_16x16x4_f32(false, a, false, b,
                                                    (short)0, acc, false, false);
    }
    // D layout: lane -> N=r, vgpr i -> M = i + 8*h
    #pragma unroll
    for (int i = 0; i < 8; ++i)
        pq[(size_t)(b0 + i + 8*h) * ATT_ + n0 + r] = acc[i];
}

// ---------------------------------------------------------------------------
// Kernel 2: fused conv+ploc GEMM (WMMA, K=64) + tanh + Wv dot -> energies
// grid: (B, T/128); block 256 = 8 waves; wave w owns t-rows [w*16, w*16+16)
// Weff staged in LDS as interleaved (even-j, odd-j) pairs so each B-fragment
// fetch is a single ds_load_b64. processed_memory streamed with NT policy.
// ---------------------------------------------------------------------------
__global__ __launch_bounds__(256) void energy_kernel(
    const float* __restrict__ aw,     // [B,2,T]
    const float* __restrict__ pm,     // [B,T,ATT]
    const float* __restrict__ pq,     // [B,ATT]
    const float* __restrict__ weff,   // [64,128]
    const float* __restrict__ Wv,     // [128]
    float* __restrict__ e)            // [B,T]
{
    __shared__ v2f   wpairLds[(KJP_/2) * WPSTRIDE_];  // 32 rows of pairs, ~34.8 KB
    __shared__ float awLds[2 * (TCHUNK_ + 30)];       // halo window, 2*158
    __shared__ float pqLds[ATT_];
    __shared__ float wvLds[ATT_];

    const int tid = threadIdx.x;
    const int b   = blockIdx.x;
    const int t0  = blockIdx.y * TCHUNK_;

    // stage Weff as (Weff[2jp][a], Weff[2jp+1][a]) pairs
    for (int idx = tid; idx < (KJP_/2)*ATT_; idx += 256) {
        int jp = idx >> 7, a = idx & (ATT_-1);
        v2f p;
        p.x = weff[(2*jp)    * ATT_ + a];
        p.y = weff[(2*jp + 1)* ATT_ + a];
        wpairLds[jp*WPSTRIDE_ + a] = p;
    }
    for (int idx = tid; idx < 2*158; idx += 256) {
        int c = idx / 158, o = idx - c*158;
        int gt = t0 + o - 15;
        awLds[idx] = (gt >= 0 && gt < T_) ? aw[((size_t)b*2 + c)*T_ + gt] : 0.f;
    }
    if (tid < ATT_) { pqLds[tid] = pq[(size_t)b*ATT_ + tid]; wvLds[tid] = Wv[tid]; }
    __syncthreads();

    const int wave = tid >> 5, lane = tid & 31;
    const int h = lane >> 4, r = lane & 15;
    const int tw0 = wave * 16;
    const int tRow = tw0 + r;

    // Preload A fragments (im2row of aw) for all 16 K-steps into registers.
    v2f af[16];
    #pragma unroll
    for (int s = 0; s < 16; ++s) {
        int j0 = 4*s + 2*h;
        float x0 = 0.f, x1 = 0.f;
        if (j0 < KJ_)   { int c = j0/KS_,     k = j0 - c*KS_;     x0 = awLds[c*158 + tRow + k]; }
        if (j0+1 < KJ_) { int c = (j0+1)/KS_, k = (j0+1) - c*KS_; x1 = awLds[c*158 + tRow + k]; }
        af[s].x = x0; af[s].y = x1;
    }

    float epart[8];
    #pragma unroll
    for (int i = 0; i < 8; ++i) epart[i] = 0.f;

    const size_t pmBase = ((size_t)b * T_ + t0) * ATT_;
    for (int at = 0; at < 8; ++at) {           // 8 a-tiles cover ATT_=128
        const int col = at*16 + r;
        v8f acc = {0,0,0,0,0,0,0,0};
        #pragma unroll
        for (int s = 0; s < 16; ++s) {
            // B-fragment pair (K = 4s+2h, 4s+2h+1) with one ds_load_b64
            v2f bf = wpairLds[(2*s + h)*WPSTRIDE_ + col];
            acc = __builtin_amdgcn_wmma_f32_16x16x4_f32(false, af[s], false, bf,
                                                        (short)0, acc, false, false);
        }
        // fused epilogue: stream processed_memory once (NT), tanh, weight by Wv
        #pragma unroll
        for (int i = 0; i < 8; ++i) {
            int tl = tw0 + i + 8*h;            // D row = i + 8*h
            float pmv = __builtin_nontemporal_load(pm + pmBase + (size_t)tl*ATT_ + col);
            float x = acc[i] + pqLds[col] + pmv;
            epart[i] += wvLds[col] * tanhf(x);
        }
    }
    // reduce over the 16 lanes of each half (each half owns disjoint t rows)
    #pragma unroll
    for (int i = 0; i < 8; ++i) {
        float v = epart[i];
        v += __shfl_xor(v, 1, 16);
        v += __shfl_xor(v, 2, 16);
        v += __shfl_xor(v, 4, 16);
        v += __shfl_xor(v, 8, 16);
        if (r == 0) e[(size_t)b*T_ + t0 + tw0 + i + 8*h] = v;
    }
}

// ---------------------------------------------------------------------------
// Kernel 3: masked softmax over T per batch row
// ---------------------------------------------------------------------------
__global__ __launch_bounds__(256) void softmax_kernel(const float* __restrict__ e,
                                                      const unsigned char* __restrict__ mask,
                                                      float* __restrict__ w)
{
    __shared__ float red[256];
    const int b = blockIdx.x, tid = threadIdx.x;
    float v[8];
    float m = -INFINITY;
    #pragma unroll
    for (int i = 0; i < 8; ++i) {
        int t = tid + i*256;
        float x = e[(size_t)b*T_ + t];
        if (mask[(size_t)b*T_ + t]) x = -INFINITY;
        v[i] = x; m = fmaxf(m, x);
    }
    red[tid] = m; __syncthreads();
    for (int s = 128; s > 0; s >>= 1) {
        if (tid < s) red[tid] = fmaxf(red[tid], red[tid+s]);
        __syncthreads();
    }
    m = red[0]; __syncthreads();
    float sum = 0.f;
    #pragma unroll
    for (int i = 0; i < 8; ++i) { v[i] = __expf(v[i] - m); sum += v[i]; }
    red[tid] = sum; __syncthreads();
    for (int s = 128; s > 0; s >>= 1) {
        if (tid < s) red[tid] += red[tid+s];
        __syncthreads();
    }
    float inv = 1.f / red[0];
    #pragma unroll
    for (int i = 0; i < 8; ++i)
        w[(size_t)b*T_ + tid + i*256] = v[i] * inv;
}

// ---------------------------------------------------------------------------
// Kernel 4: partial context  ctx_part[sp,b,d] = sum_{t in split} w[b,t]*mem[b,t,d]
// Pure 1-GiB stream of `memory` with NT loads, 4-way T split for occupancy.
// ---------------------------------------------------------------------------
__global__ __launch_bounds__(256) void context_partial_kernel(
    const float* __restrict__ mem,    // [B,T,EMB]
    const float* __restrict__ w,      // [B,T]
    float* __restrict__ part)         // [NSPLIT,B,EMB]
{
    __shared__ float wLds[T_ / NSPLIT_];   // 512 weights
    const int b = blockIdx.x, sp = blockIdx.y, tid = threadIdx.x;
    const int t0 = sp * (T_ / NSPLIT_);
    for (int i = tid; i < T_/NSPLIT_; i += 256)
        wLds[i] = w[(size_t)b*T_ + t0 + i];
    __syncthreads();

    const v2f* m2 = (const v2f*)(mem + ((size_t)b*T_ + t0) * EMB_);
    v2f acc = {0.f, 0.f};
    for (int t = 0; t < T_/NSPLIT_; ++t) {
        float wt = wLds[t];
        v2f mv = __builtin_nontemporal_load(m2 + (size_t)t*(EMB_/2) + tid);
        acc.x += wt * mv.x; acc.y += wt * mv.y;
    }
    float* dst = part + ((size_t)sp*B_ + b) * EMB_ + 2*tid;
    dst[0] = acc.x; dst[1] = acc.y;
}

// ---------------------------------------------------------------------------
// Kernel 5: deterministic reduction of the 4 partials into d_out context
// ---------------------------------------------------------------------------
__global__ void context_reduce_kernel(const float* __restrict__ part,
                                      float* __restrict__ out)
{
    int idx = blockIdx.x * 256 + threadIdx.x;   // 0 .. B*EMB-1
    float s = 0.f;
    #pragma unroll
    for (int sp = 0; sp < NSPLIT_; ++sp)
        s += part[(size_t)sp*B_*EMB_ + idx];
    out[idx] = s;
}

// ---------------------------------------------------------------------------
extern "C" void kernel_launch(void* const* d_in, const int* in_sizes, int n_in,
                              void* d_out, int out_size, void* d_ws, size_t ws_size,
                              hipStream_t stream)
{
    const float* H     = (const float*)d_in[0];   // [256,1024]
    const float* mem   = (const float*)d_in[1];   // [256,2048,512]
    const float* pm    = (const float*)d_in[2];   // [256,2048,128]
    const float* aw    = (const float*)d_in[3];   // [256,2,2048]
    const unsigned char* mask = (const unsigned char*)d_in[4]; // [256,2048] bool
    const float* Wq    = (const float*)d_in[5];   // [128,1024]
    const float* convw = (const float*)d_in[6];   // [32,2,31]
    const float* Wloc  = (const float*)d_in[7];   // [128,32]
    const float* Wv    = (const float*)d_in[8];   // [1,128]

    float* out  = (float*)d_out;
    float* ws   = (float*)d_ws;
    float* pq   = ws + WS_PQ;
    float* weff = ws + WS_WEFF;
    float* e    = ws + WS_E;
    float* part = ws + WS_PART;
    float* ctx  = out;                 // [256,512]  (output 0)
    float* attw = out + B_*EMB_;       // [256,2048] (output 1)

    weff_kernel<<<1, 256, 0, stream>>>(convw, Wloc, weff);
    pq_kernel<<<B_/16, 256, 0, stream>>>(H, Wq, pq);
    energy_kernel<<<dim3(B_, T_/TCHUNK_), 256, 0, stream>>>(aw, pm, pq, weff, Wv, e);
    softmax_kernel<<<B_, 256, 0, stream>>>(e, mask, attw);
    context_partial_kernel<<<dim3(B_, NSPLIT_), 256, 0, stream>>>(mem, attw, part);
    context_reduce_kernel<<<(B_*EMB_)/256, 256, 0, stream>>>(part, ctx);
}